// PolylineSubgraphEncoder_21397527068864
// MI455X (gfx1250) — compile-verified
//
#include <hip/hip_runtime.h>

typedef __attribute__((ext_vector_type(2))) float v2f;
typedef __attribute__((ext_vector_type(8))) float v8f;

#define HID 64

// ---------------- degree / normalization ----------------

__global__ void zero_deg_kernel(int* __restrict__ deg, int N) {
  int i = blockIdx.x * blockDim.x + threadIdx.x;
  if (i < N) deg[i] = 0;
}

__global__ void count_deg_kernel(const long long* __restrict__ dst, int* __restrict__ deg, int E) {
  int e = blockIdx.x * blockDim.x + threadIdx.x;
  if (e < E) atomicAdd(&deg[(int)dst[e]], 1);
}

__global__ void dinv_kernel(const int* __restrict__ deg, float* __restrict__ dinv, int N) {
  int i = blockIdx.x * blockDim.x + threadIdx.x;
  if (i < N) dinv[i] = rsqrtf((float)(deg[i] + 1));  // +1 self-loop
}

// ---------------- layer 1 GEMM: g = (x @ W1) * dinv[row]; acc = g ----------------
// One wave32 per 16-row tile. K = 4 -> single V_WMMA_F32_16X16X4_F32 per 16x16 n-tile.

__global__ void gcn_gemm1_wmma(const float* __restrict__ x, const float* __restrict__ W,
                               const float* __restrict__ dinv, float* __restrict__ g,
                               float* __restrict__ acc, int N) {
  int wave = (int)((blockIdx.x * blockDim.x + threadIdx.x) >> 5);
  int lane = threadIdx.x & 31;
  int tiles = (N + 15) >> 4;
  if (wave >= tiles) return;              // wave-uniform: EXEC stays all-ones for WMMA
  int ln = lane & 15, hi = lane >> 4;
  int row0 = wave << 4;
  int row  = row0 + ln;
  int rowc = row < N ? row : N - 1;

  // A 16x4 f32: lane ln holds A[ln][2*hi], A[ln][2*hi+1]
  v2f a = *(const v2f*)(x + (size_t)rowc * 4 + 2 * hi);

  v8f c[4];
#pragma unroll
  for (int nt = 0; nt < 4; ++nt) {
    v2f b;
    b.x = W[(2 * hi)     * HID + nt * 16 + ln];
    b.y = W[(2 * hi + 1) * HID + nt * 16 + ln];
    v8f z = {};
    c[nt] = __builtin_amdgcn_wmma_f32_16x16x4_f32(false, a, false, b, (short)0, z, false, false);
  }

#pragma unroll
  for (int r = 0; r < 8; ++r) {
    int m = row0 + r + 8 * hi;            // D: vgpr r = row (r + 8*hi), col = ln
    if (m < N) {
      float dv = dinv[m];
#pragma unroll
      for (int nt = 0; nt < 4; ++nt) {
        float v = c[nt][r] * dv;
        size_t o = (size_t)m * HID + nt * 16 + ln;
        g[o]   = v;
        acc[o] = v;                       // self-loop init: acc starts at g[d]
      }
    }
  }
}

// ---------------- layer 2 GEMM: g = (h @ W2) * dinv[row]; acc = g ----------------
// W2 (64x64) staged in LDS with padded stride 65 (no bank conflict between half-waves).
// 16 k-steps x 4 n-tiles = 64 WMMAs per wave.

__global__ void gcn_gemm2_wmma(const float* __restrict__ h, const float* __restrict__ W,
                               const float* __restrict__ dinv, float* __restrict__ g,
                               float* __restrict__ acc, int N) {
  __shared__ float ws[64 * 65];
  int tid = threadIdx.x;
#pragma unroll
  for (int i = 0; i < 16; ++i) {
    int idx = tid + i * 256;              // 4096 floats / 256 threads
    ws[(idx >> 6) * 65 + (idx & 63)] = W[idx];
  }
  __syncthreads();

  int wave = (int)((blockIdx.x * blockDim.x + tid) >> 5);
  int lane = tid & 31;
  int tiles = (N + 15) >> 4;
  if (wave >= tiles) return;              // wave-uniform
  int ln = lane & 15, hi = lane >> 4;
  int row0 = wave << 4;
  int row  = row0 + ln;
  int rowc = row < N ? row : N - 1;
  const float* hrow = h + (size_t)rowc * HID;

  v8f c[4] = {v8f{}, v8f{}, v8f{}, v8f{}};
#pragma unroll
  for (int kk = 0; kk < 16; ++kk) {
    int k = 4 * kk + 2 * hi;
    v2f a = *(const v2f*)(hrow + k);      // A[ln][k], A[ln][k+1]
#pragma unroll
    for (int nt = 0; nt < 4; ++nt) {
      v2f b;
      b.x = ws[k       * 65 + nt * 16 + ln];
      b.y = ws[(k + 1) * 65 + nt * 16 + ln];
      c[nt] = __builtin_amdgcn_wmma_f32_16x16x4_f32(false, a, false, b, (short)0, c[nt], false, false);
    }
  }

#pragma unroll
  for (int r = 0; r < 8; ++r) {
    int m = row0 + r + 8 * hi;
    if (m < N) {
      float dv = dinv[m];
#pragma unroll
      for (int nt = 0; nt < 4; ++nt) {
        float v = c[nt][r] * dv;
        size_t o = (size_t)m * HID + nt * 16 + ln;
        g[o]   = v;
        acc[o] = v;
      }
    }
  }
}

// ---------------- edge scatter: acc[dst] += g[src], 16 lanes/edge, float4 each ----------------

__global__ void scatter_kernel(const float* __restrict__ g, float* __restrict__ acc,
                               const long long* __restrict__ src, const long long* __restrict__ dst,
                               int E) {
  long long t = (long long)blockIdx.x * blockDim.x + threadIdx.x;
  int e = (int)(t >> 4);
  if (e >= E) return;
  int c = ((int)t & 15) << 2;
  int s = (int)src[e];
  int d = (int)dst[e];
  float4 v = *(const float4*)(g + (size_t)s * HID + c);
  float* p = acc + (size_t)d * HID + c;
  atomicAdd(p + 0, v.x);
  atomicAdd(p + 1, v.y);
  atomicAdd(p + 2, v.z);
  atomicAdd(p + 3, v.w);
}

// ---------------- epilogue: out = relu(acc * dinv[node] + bias) ----------------

__global__ void epilogue_kernel(const float* __restrict__ acc, const float* __restrict__ dinv,
                                const float* __restrict__ bias, float* __restrict__ out, int N) {
  int i = blockIdx.x * blockDim.x + threadIdx.x;
  int total = N * (HID / 4);
  if (i >= total) return;
  int node = i >> 4;
  int c = (i & 15) << 2;
  float dv = dinv[node];
  float4 v = *(const float4*)(acc + (size_t)node * HID + c);
  float4 b = *(const float4*)(bias + c);
  float4 r;
  r.x = fmaxf(fmaf(v.x, dv, b.x), 0.f);
  r.y = fmaxf(fmaf(v.y, dv, b.y), 0.f);
  r.z = fmaxf(fmaf(v.z, dv, b.z), 0.f);
  r.w = fmaxf(fmaf(v.w, dv, b.w), 0.f);
  *(float4*)(out + (size_t)node * HID + c) = r;
}

// ---------------- launch ----------------

extern "C" void kernel_launch(void* const* d_in, const int* in_sizes, int n_in,
                              void* d_out, int out_size, void* d_ws, size_t ws_size,
                              hipStream_t stream) {
  const float*     x  = (const float*)d_in[0];
  const long long* ei = (const long long*)d_in[1];   // edge_index: int64 per reference
  const float*     W1 = (const float*)d_in[2];
  const float*     b1 = (const float*)d_in[3];
  const float*     W2 = (const float*)d_in[4];
  const float*     b2 = (const float*)d_in[5];

  int N = in_sizes[0] / 4;     // IN_DIM = 4
  int E = in_sizes[1] / 2;
  const long long* src = ei;
  const long long* dst = ei + E;

  // workspace carve-out
  char* ws = (char*)d_ws;
  size_t off = 0;
  auto carve = [&](size_t bytes) {
    void* p = ws + off;
    off += (bytes + 255) & ~(size_t)255;
    return p;
  };
  int*   deg  = (int*)  carve((size_t)N * 4);
  float* dinv = (float*)carve((size_t)N * 4);
  float* B0   = (float*)carve((size_t)N * HID * 4);  // g
  float* B1   = (float*)carve((size_t)N * HID * 4);  // acc
  float* B2   = (float*)carve((size_t)N * HID * 4);  // h1
  (void)ws_size; (void)n_in; (void)out_size;

  const int T = 256;
  int nb_nodes = (N + T - 1) / T;
  int nb_edges = (E + T - 1) / T;
  int tiles    = (N + 15) / 16;
  int nb_gemm  = (tiles * 32 + T - 1) / T;
  long long sc_threads = (long long)E * 16;
  int nb_scat  = (int)((sc_threads + T - 1) / T);
  int nb_epi   = (N * (HID / 4) + T - 1) / T;

  // degrees + symmetric normalization
  zero_deg_kernel<<<nb_nodes, T, 0, stream>>>(deg, N);
  count_deg_kernel<<<nb_edges, T, 0, stream>>>(dst, deg, E);
  dinv_kernel<<<nb_nodes, T, 0, stream>>>(deg, dinv, N);

  // layer 1
  gcn_gemm1_wmma<<<nb_gemm, T, 0, stream>>>(x, W1, dinv, B0, B1, N);
  scatter_kernel<<<nb_scat, T, 0, stream>>>(B0, B1, src, dst, E);
  epilogue_kernel<<<nb_epi, T, 0, stream>>>(B1, dinv, b1, B2, N);

  // layer 2
  gcn_gemm2_wmma<<<nb_gemm, T, 0, stream>>>(B2, W2, dinv, B0, B1, N);
  scatter_kernel<<<nb_scat, T, 0, stream>>>(B0, B1, src, dst, E);
  epilogue_kernel<<<nb_epi, T, 0, stream>>>(B1, dinv, b2, (float*)d_out, N);
}